// PatchCore_80032420594277
// MI455X (gfx1250) — compile-verified
//
#include <hip/hip_runtime.h>

// PatchCore greedy coreset subsampling for MI455X (gfx1250, wave32).
//  N=16384 patches, D_IN=1024, D_PROJ=128, 163 greedy iterations.
// Phases:
//  1) proj:    reduced = features @ W                (f32 WMMA 16x16x4)
//  2) sq:      per-row squared norms of reduced
//  3) gram:    row-sums of clipped dist^2 over the 16384^2 Gram matrix,
//              fused (dist never materialized; reduced is L2-resident).
//              8 column-chunks per row-strip -> 8192 waves, private partial
//              slots (no FP atomics -> bitwise deterministic).
//  4) combine: anchorSq[i] = sum of 8 partials (fixed order)
//  5) greedy:  persistent 1-block kernel, anchor in registers, shuffle argmax
//  6) gather:  d_out = features[sel]

#define NPTS   16384
#define DIN    1024
#define DPROJ  128
#define NSAMP  163
#define NCHUNK 8                       // column chunks in gram phase
#define TILES_PER_CHUNK (NPTS / 16 / NCHUNK)   // 128

typedef float v2f __attribute__((ext_vector_type(2)));
typedef float v8f __attribute__((ext_vector_type(8)));

__device__ __forceinline__ v8f wmma_f32_k4(v2f a, v2f b, v8f c) {
  // V_WMMA_F32_16X16X4_F32: D = A(16x4) * B(4x16) + C(16x16), full f32.
  return __builtin_amdgcn_wmma_f32_16x16x4_f32(
      /*neg_a=*/false, a, /*neg_b=*/false, b,
      /*c_mod=*/(short)0, c, /*reuse_a=*/false, /*reuse_b=*/false);
}

// ---------------------------------------------------------------- phase 1
// One wave per 16x16 output tile of reduced[N, DPROJ]. 8 waves / block.
__global__ __launch_bounds__(256)
void proj_kernel(const float* __restrict__ F, const float* __restrict__ W,
                 float* __restrict__ R) {
  const int wave = (blockIdx.x * blockDim.x + threadIdx.x) >> 5;
  const int lane = threadIdx.x & 31;
  const int tm = wave >> 3;          // DPROJ/16 = 8 tiles in N-dim
  const int tn = wave & 7;
  const int m0 = tm * 16, n0 = tn * 16;
  const int half = lane >> 4;        // 0: K{0,1}, C rows 0-7; 1: K{2,3}
  const int r16  = lane & 15;
  const int koff = half * 2;

  v8f acc = {};
  const float* arow = F + (size_t)(m0 + r16) * DIN + koff;
  for (int k = 0; k < DIN; k += 4) {
    v2f a = *(const v2f*)(arow + k);                    // A[M=r16, K{0,1}]
    v2f b;
    b.x = W[(size_t)(k + koff)     * DPROJ + n0 + r16]; // B[K, N=r16]
    b.y = W[(size_t)(k + koff + 1) * DPROJ + n0 + r16];
    acc = wmma_f32_k4(a, b, acc);
  }
#pragma unroll
  for (int i = 0; i < 8; i++)
    R[(size_t)(m0 + i + half * 8) * DPROJ + n0 + r16] = acc[i];
}

// ---------------------------------------------------------------- phase 2
// One wave per row: sq[i] = ||reduced[i,:]||^2
__global__ __launch_bounds__(256)
void sq_kernel(const float* __restrict__ R, float* __restrict__ sq) {
  const int row  = (blockIdx.x * blockDim.x + threadIdx.x) >> 5;
  const int lane = threadIdx.x & 31;
  const float* rp = R + (size_t)row * DPROJ;
  float s = 0.f;
#pragma unroll
  for (int k = lane; k < DPROJ; k += 32) { float v = rp[k]; s += v * v; }
#pragma unroll
  for (int m = 16; m >= 1; m >>= 1) s += __shfl_xor(s, m, 32);
  if (lane == 0) sq[row] = s;
}

// ---------------------------------------------------------------- phase 3
// wave w: strip = w>>3 (16 rows), chunk = w&7 (128 column tiles).
// A fragments preloaded (32 x v2f = 64 VGPRs); per-row partial sums of
// clipped dist^2 in registers; 16-lane shuffle reduce; store to a private
// partial slot (written exactly once -> no init, no atomics, deterministic).
__global__ __launch_bounds__(256)
void gram_rowsum_kernel(const float* __restrict__ R, const float* __restrict__ sq,
                        float* __restrict__ anchorPart) {
  const int wave  = (blockIdx.x * blockDim.x + threadIdx.x) >> 5; // 0..8191
  const int lane  = threadIdx.x & 31;
  const int chunk = wave & (NCHUNK - 1);
  const int strip = wave >> 3;
  const int half  = lane >> 4;
  const int r16   = lane & 15;
  const int koff  = half * 2;
  const int m0    = strip * 16;

  v2f afrag[32];
  const float* arow = R + (size_t)(m0 + r16) * DPROJ + koff;
#pragma unroll
  for (int kk = 0; kk < 32; kk++) afrag[kk] = *(const v2f*)(arow + kk * 4);

  float sqr[8];
#pragma unroll
  for (int i = 0; i < 8; i++) sqr[i] = sq[m0 + i + half * 8];

  float partial[8] = {0.f, 0.f, 0.f, 0.f, 0.f, 0.f, 0.f, 0.f};

  const int t0 = chunk * TILES_PER_CHUNK;
  for (int t = t0; t < t0 + TILES_PER_CHUNK; t++) {
    const int col = t * 16 + r16;
    const float sqc = sq[col];
    const float* brow = R + (size_t)col * DPROJ + koff;     // B[k,j] = R[j,k]
    // hide L2 latency of the next tile behind this tile's 32-WMMA burst
    __builtin_prefetch((const void*)(brow + 16 * DPROJ), 0, 1);
    v8f acc = {};
#pragma unroll
    for (int kk = 0; kk < 32; kk++) {
      v2f b = *(const v2f*)(brow + kk * 4);
      acc = wmma_f32_k4(afrag[kk], b, acc);
    }
#pragma unroll
    for (int i = 0; i < 8; i++)
      partial[i] += fmaxf(0.f, sqr[i] + sqc - 2.f * acc[i]);
  }

#pragma unroll
  for (int i = 0; i < 8; i++) {
    float p = partial[i];
    p += __shfl_xor(p, 1, 32);
    p += __shfl_xor(p, 2, 32);
    p += __shfl_xor(p, 4, 32);
    p += __shfl_xor(p, 8, 32);       // reduce within each 16-lane half
    if (r16 == 0)
      anchorPart[(size_t)chunk * NPTS + m0 + i + half * 8] = p;
  }
}

// ---------------------------------------------------------------- phase 4
// Fixed-order combine of the 8 chunk partials -> bitwise deterministic.
__global__ __launch_bounds__(256)
void combine_kernel(const float* __restrict__ anchorPart,
                    float* __restrict__ anchorSq) {
  const int i = blockIdx.x * 256 + threadIdx.x;
  float s = 0.f;
#pragma unroll
  for (int c = 0; c < NCHUNK; c++) s += anchorPart[(size_t)c * NPTS + i];
  anchorSq[i] = s;
}

// ---------------------------------------------------------------- phase 5
// Persistent greedy farthest-point sampling: one block, 1024 threads (32
// wave32s on one WGP). anchor lives in registers (16 values/thread).
// Argmax: shfl_xor butterfly per wave (value,index with min-index tiebreak:
// commutative+associative, order-safe), then one wave reduces 32 winners.
// Only 4 __syncthreads per iteration.
__global__ __launch_bounds__(1024)
void greedy_kernel(const float* __restrict__ R, const float* __restrict__ sq,
                   const float* __restrict__ anchorSq, int* __restrict__ sel) {
  __shared__ float redV[32];
  __shared__ int   redI[32];
  __shared__ float selRow[DPROJ];
  __shared__ float selSqS;
  __shared__ int   selIdxS;

  const int t    = threadIdx.x;
  const int lane = t & 31;
  const int wid  = t >> 5;

  float aVal[16], sqOwn[16];
#pragma unroll
  for (int j = 0; j < 16; j++) {
    const int row = t + j * 1024;
    aVal[j]  = sqrtf(anchorSq[row]);   // anchor0 = sqrt(sum_j dist^2)
    sqOwn[j] = sq[row];
  }

  for (int s = 0; s < NSAMP; s++) {
    // ---- local argmax over this thread's 16 rows (ties -> lowest index)
    float bv = aVal[0]; int bi = t;
#pragma unroll
    for (int j = 1; j < 16; j++)
      if (aVal[j] > bv) { bv = aVal[j]; bi = t + j * 1024; }
    // ---- wave32 butterfly
#pragma unroll
    for (int m = 1; m < 32; m <<= 1) {
      const float ov = __shfl_xor(bv, m, 32);
      const int   oi = __shfl_xor(bi, m, 32);
      if (ov > bv || (ov == bv && oi < bi)) { bv = ov; bi = oi; }
    }
    if (lane == 0) { redV[wid] = bv; redI[wid] = bi; }
    __syncthreads();
    // ---- wave 0 reduces the 32 per-wave winners
    if (wid == 0) {
      bv = redV[lane]; bi = redI[lane];
#pragma unroll
      for (int m = 1; m < 32; m <<= 1) {
        const float ov = __shfl_xor(bv, m, 32);
        const int   oi = __shfl_xor(bi, m, 32);
        if (ov > bv || (ov == bv && oi < bi)) { bv = ov; bi = oi; }
      }
      if (lane == 0) { selIdxS = bi; selSqS = sq[bi]; sel[s] = bi; }
    }
    __syncthreads();
    const int si = selIdxS;
    if (t < DPROJ) selRow[t] = R[(size_t)si * DPROJ + t];
    __syncthreads();

    // ---- anchor = min(anchor, dist(:, si)); column recomputed from R (L2)
    const float selSq = selSqS;
#pragma unroll 4
    for (int j = 0; j < 16; j++) {
      const int row = t + j * 1024;
      const float4* rp = (const float4*)(R + (size_t)row * DPROJ);
      const float4* wp = (const float4*)selRow;
      float dot = 0.f;
#pragma unroll
      for (int k = 0; k < DPROJ / 4; k++) {
        const float4 v = rp[k];
        const float4 w = wp[k];
        dot += v.x * w.x + v.y * w.y + v.z * w.z + v.w * w.w;
      }
      const float d2 = sqOwn[j] + selSq - 2.f * dot;
      aVal[j] = fminf(aVal[j], sqrtf(fmaxf(d2, 0.f)));
    }
    // selRow/selIdxS are rewritten only after the first barrier of the next
    // iteration, which follows this implicit join point:
    __syncthreads();
  }
}

// ---------------------------------------------------------------- phase 6
__global__ __launch_bounds__(256)
void gather_kernel(const float* __restrict__ F, const int* __restrict__ sel,
                   float* __restrict__ out) {
  const int s = blockIdx.x;
  const int c = blockIdx.y * 256 + threadIdx.x;
  out[(size_t)s * DIN + c] = F[(size_t)sel[s] * DIN + c];
}

// ----------------------------------------------------------------
extern "C" void kernel_launch(void* const* d_in, const int* in_sizes, int n_in,
                              void* d_out, int out_size, void* d_ws, size_t ws_size,
                              hipStream_t stream) {
  (void)in_sizes; (void)n_in; (void)out_size; (void)ws_size;
  const float* F = (const float*)d_in[0];   // [16384, 1024]
  const float* W = (const float*)d_in[1];   // [1024, 128]

  float* R          = (float*)d_ws;                       // [N, DPROJ] 8 MB
  float* sq         = R + (size_t)NPTS * DPROJ;           // [N]
  float* anchorSq   = sq + NPTS;                          // [N]
  float* anchorPart = anchorSq + NPTS;                    // [NCHUNK, N] 512 KB
  int*   sel        = (int*)(anchorPart + (size_t)NCHUNK * NPTS); // [NSAMP]

  // 8192 tiles of reduced, 8 waves/block
  proj_kernel<<<1024, 256, 0, stream>>>(F, W, R);
  // one wave per row, 8 rows/block
  sq_kernel<<<NPTS / 8, 256, 0, stream>>>(R, sq);
  // 1024 row-strips x 8 column-chunks = 8192 waves, 8 waves/block
  gram_rowsum_kernel<<<1024, 256, 0, stream>>>(R, sq, anchorPart);
  // fixed-order partial combine
  combine_kernel<<<NPTS / 256, 256, 0, stream>>>(anchorPart, anchorSq);
  // single persistent workgroup for the 163 serial iterations
  greedy_kernel<<<1, 1024, 0, stream>>>(R, sq, anchorSq, sel);
  // output gather
  gather_kernel<<<dim3(NSAMP, DIN / 256), 256, 0, stream>>>(F, sel, (float*)d_out);
}